// KPConv_15668040696079
// MI455X (gfx1250) — compile-verified
//
#include <hip/hip_runtime.h>

// ---------------------------------------------------------------------------
// KPConv for MI455X (gfx1250, wave32).
// f32 WMMA (V_WMMA_F32_16X16X4_F32) for both matmuls (problem is gather/L2-
// bound: ~10 GFLOP vs 60MB HBM / ~1.3GB L2 traffic), plus async-to-LDS
// double buffering of the 960x128 weight matrix in stage B.
// ---------------------------------------------------------------------------

typedef float v2f __attribute__((ext_vector_type(2)));
typedef float v4f __attribute__((ext_vector_type(4)));
typedef float v8f __attribute__((ext_vector_type(8)));
typedef int   v4i __attribute__((ext_vector_type(4)));

// Address-space-qualified int4 pointee types for the async builtin
// (param 1 is 'int4 __device__*' = AS1, param 2 is LDS = AS3).
typedef __attribute__((address_space(1))) v4i gv4i;
typedef __attribute__((address_space(3))) v4i lv4i;

#define BN   2
#define NN   65536
#define MM   16384
#define KK   32
#define KP   15
#define DIN  64
#define DOUT 128

#define WSTRIDE 962   // s_weighted row stride (dwords): even (8B align) and
                      // 962 % 64 == 2 -> 16 lanes hit 16 distinct LDS banks

#if __has_builtin(__builtin_amdgcn_global_load_async_to_lds_b128) && \
    __has_builtin(__builtin_amdgcn_s_wait_asynccnt)
#define USE_ASYNC 1
#else
#define USE_ASYNC 0
#endif

// ---------------------------------------------------------------------------
// Kernel 1: transpose features (B, Din, N) -> featT (B, N+1, Din), shadow row 0
// ---------------------------------------------------------------------------
__global__ __launch_bounds__(256) void kpconv_transpose(
    const float* __restrict__ features, float* __restrict__ featT) {
  __shared__ float tile[64][65];
  const int tilesPerB = (NN / 64) + 1;                 // 1025 (last tile: shadow row)
  const int b  = blockIdx.x / tilesPerB;
  const int n0 = (blockIdx.x % tilesPerB) * 64;
  const int t  = threadIdx.x;

#pragma unroll
  for (int i = 0; i < 16; ++i) {
    int e  = t + i * 256;
    int d  = e >> 6;
    int nl = e & 63;
    int n  = n0 + nl;
    float v = 0.0f;
    if (n < NN) v = features[((size_t)b * DIN + d) * NN + n];  // coalesced in n
    tile[nl][d] = v;
  }
  __syncthreads();
#pragma unroll
  for (int i = 0; i < 16; ++i) {
    int e  = t + i * 256;
    int nl = e >> 6;
    int d  = e & 63;
    int n  = n0 + nl;
    if (n < NN + 1)
      featT[((size_t)b * (NN + 1) + n) * DIN + d] = tile[nl][d];  // coalesced in d
  }
}

// ---------------------------------------------------------------------------
// Async chunk issue: 64x128 f32 (32 KB) of weights into an LDS buffer.
// 8 x b128 per thread, each lane writes 16B to LDS (ASYNCcnt += 8 per wave).
// ---------------------------------------------------------------------------
#if USE_ASYNC
__device__ __forceinline__ void issue_chunk_async(const float* wsrc,
                                                  float* ldst, int tid) {
  float* ws = const_cast<float*>(wsrc);
#pragma unroll
  for (int i = 0; i < 8; ++i) {
    int flat = tid * 4 + i * 1024;
    __builtin_amdgcn_global_load_async_to_lds_b128(
        (gv4i*)(ws + flat), (lv4i*)(ldst + flat), 0, 0);
  }
}
#endif

// ---------------------------------------------------------------------------
// Kernel 2: fused geometry + per-center matmul + big GEMM.
// One 256-thread block (8 wave32) handles 16 centers.
// ---------------------------------------------------------------------------
__global__ __launch_bounds__(256) void kpconv_main(
    const float* __restrict__ points_xyz,
    const float* __restrict__ center_xyz,
    const long long* __restrict__ nbr_idx,
    const float* __restrict__ kernel_points,
    const float* __restrict__ weights,       // (Kp, Din, Dout) = (960 x 128) flat
    const float* __restrict__ featT,
    float* __restrict__ out) {               // (B, Dout, M)

  __shared__ int   s_idx[16][KK];                 // neighbor row BYTE offsets
  __shared__ float s_nxyz[16][KK][3];
  __shared__ float s_dist[16][KK];
  __shared__ float s_mdist[16];
  __shared__ float s_cxyz[16][3];
  __shared__ float s_kp[KP][3];
  __shared__ float s_u[64 * DOUT];                // 32 KB: stepA w^T / stepB W-chunk (odd)
  __shared__ float s_wbuf[64 * DOUT];             // 32 KB: stepB W-chunk (even)
  __shared__ float s_weighted[16 * WSTRIDE];      // ~60 KB: weighted[m][pd], pd=p*64+d

  const int tid  = threadIdx.x;
  const int lane = tid & 31;
  const int wave = tid >> 5;      // 0..7
  const int col  = lane & 15;     // matrix column within tile
  const int hi   = lane >> 4;     // lane half (selects K sub-pair in f32 WMMA)

  const int tilesPerB = MM / 16;  // 1024
  const int b  = blockIdx.x / tilesPerB;
  const int m0 = (blockIdx.x % tilesPerB) * 16;

#if USE_ASYNC
  // Prefetch weight chunk 0 under the whole geometry + stage-A phase.
  issue_chunk_async(weights, s_wbuf, tid);
#endif

  // ---- stage 0: per-tile constants ---------------------------------------
  if (tid < KP * 3) s_kp[tid / 3][tid % 3] = kernel_points[tid];
  if (tid < 48)
    s_cxyz[tid / 3][tid % 3] =
        center_xyz[((size_t)b * MM + m0 + tid / 3) * 3 + (tid % 3)];
  __syncthreads();

  // ---- stage 1: gather neighbor xyz, distances ---------------------------
#pragma unroll
  for (int pr = 0; pr < 2; ++pr) {
    int e = tid + pr * 256;       // 512 (m,k) pairs
    int m = e >> 5;
    int k = e & 31;
    long long idx64 = nbr_idx[((size_t)b * MM + (m0 + m)) * KK + k];
    int idx = (int)idx64;
    s_idx[m][k] = idx * (DIN * 4);          // byte offset into featT slab (<=2^24)
    float px, py, pz;
    if (idx < NN) {
      const float* p = points_xyz + ((size_t)b * NN + idx) * 3;
      px = p[0]; py = p[1]; pz = p[2];
    } else {                                // shadow point
      px = -1000.0f; py = -1000.0f; pz = -1000.0f;
    }
    float nx = px - s_cxyz[m][0];
    float ny = py - s_cxyz[m][1];
    float nz = pz - s_cxyz[m][2];
    s_nxyz[m][k][0] = nx; s_nxyz[m][k][1] = ny; s_nxyz[m][k][2] = nz;
    s_dist[m][k] = sqrtf(nx * nx + ny * ny + nz * nz);
  }
  __syncthreads();
  if (tid < 16) {
    float md = 0.0f;
    for (int k = 0; k < KK; ++k) md = fmaxf(md, s_dist[tid][k]);
    s_mdist[tid] = md;
  }
  __syncthreads();

  // ---- stage 2: kernel-point weights, stored transposed w^T[m][p][k] -----
  float* s_wT = s_u;  // [16][16][KK], p padded to 16 with zeros
#pragma unroll
  for (int pr = 0; pr < 2; ++pr) {
    int e = tid + pr * 256;
    int m = e >> 5;
    int k = e & 31;
    float md = s_mdist[m];
    float inv_sigma = 1.0f / (md / 2.1f + 1e-5f);
    float nx = s_nxyz[m][k][0], ny = s_nxyz[m][k][1], nz = s_nxyz[m][k][2];
#pragma unroll
    for (int p = 0; p < KP; ++p) {
      float dx = nx - s_kp[p][0] * md;
      float dy = ny - s_kp[p][1] * md;
      float dz = nz - s_kp[p][2] * md;
      float d  = sqrtf(dx * dx + dy * dy + dz * dz);
      s_wT[(m * 16 + p) * KK + k] = fmaxf(0.0f, 1.0f - d * inv_sigma);
    }
    s_wT[(m * 16 + 15) * KK + k] = 0.0f;   // pad row
  }
  __syncthreads();

  // ---- stage A: weighted(16p x 64d) = w^T(16x32) @ feats(32x64) per center
  // f32 WMMA operand layout: lane<16 -> K sub-pair (k0,k0+1); lane>=16 -> (+2,+3).
  // Uniform base + 32-bit unsigned byte offsets -> SADDR-form global loads.
  const char* fb = (const char*)featT + (size_t)b * (NN + 1) * DIN * 4;
#pragma unroll
  for (int cc = 0; cc < 2; ++cc) {
    int m = wave + cc * 8;        // each wave owns 2 centers
#pragma unroll
    for (int dt = 0; dt < 4; ++dt) {
      int d0 = dt * 16;
      unsigned dcol4 = (unsigned)(d0 + col) * 4u;
      v8f acc = {0.f, 0.f, 0.f, 0.f, 0.f, 0.f, 0.f, 0.f};
#pragma unroll
      for (int ks = 0; ks < 8; ++ks) {
        int kA = ks * 4 + 2 * hi;
        v2f a = *(const v2f*)&s_wT[(m * 16 + col) * KK + kA];   // 8B-aligned LDS
        unsigned off0 = (unsigned)s_idx[m][kA] + dcol4;
        unsigned off1 = (unsigned)s_idx[m][kA + 1] + dcol4;
        v2f bb;
        bb.x = *(const float*)(fb + off0);   // 64B-contig per 16 lanes
        bb.y = *(const float*)(fb + off1);
        acc = __builtin_amdgcn_wmma_f32_16x16x4_f32(
            false, a, false, bb, (short)0, acc, false, false);
      }
      // D layout: VGPR v, lane L -> row (v + 8*(L/16)), col L%16
#pragma unroll
      for (int v = 0; v < 8; ++v) {
        int p = v + 8 * hi;
        s_weighted[m * WSTRIDE + p * DIN + d0 + col] = acc[v];
      }
    }
  }
  __syncthreads();                // w^T (s_u) is dead from here on

#if USE_ASYNC
  // Prefetch weight chunk 1 into the freed s_u buffer.
  issue_chunk_async(weights + (size_t)64 * DOUT, s_u, tid);
#endif

  // ---- stage B: out(16m x 128o) = weighted(16 x 960) @ W(960 x 128) ------
  const int o0 = wave * 16;       // 8 waves partition the Dout dimension
  v8f oacc = {0.f, 0.f, 0.f, 0.f, 0.f, 0.f, 0.f, 0.f};

  for (int ch = 0; ch < 15; ++ch) {
    int pd0 = ch * 64;
    float* buf = (ch & 1) ? s_u : s_wbuf;
#if USE_ASYNC
    // Retire this chunk's 8 async ops; leave the next chunk's 8 in flight.
    if (ch < 14) __builtin_amdgcn_s_wait_asynccnt(8);
    else         __builtin_amdgcn_s_wait_asynccnt(0);
#else
#pragma unroll
    for (int i = 0; i < 8; ++i) {
      int flat = tid * 4 + i * 1024;
      *(v4f*)&buf[flat] = *(const v4f*)&weights[(size_t)pd0 * DOUT + flat];
    }
#endif
    __syncthreads();              // chunk visible to all waves
#pragma unroll
    for (int ks = 0; ks < 16; ++ks) {
      int k0 = ks * 4 + 2 * hi;
      v2f a = *(const v2f*)&s_weighted[col * WSTRIDE + pd0 + k0];
      v2f bb;
      bb.x = buf[k0 * DOUT + o0 + col];
      bb.y = buf[(k0 + 1) * DOUT + o0 + col];
      oacc = __builtin_amdgcn_wmma_f32_16x16x4_f32(
          false, a, false, bb, (short)0, oacc, false, false);
    }
    __syncthreads();              // all waves done reading buf
#if USE_ASYNC
    if (ch + 2 < 15)              // refill the just-freed buffer (same parity)
      issue_chunk_async(weights + (size_t)(ch + 2) * 64 * DOUT, buf, tid);
#endif
  }

  // ---- store out[b][o][m]: each lane holds 8 consecutive m at fixed o ----
  // (rows v+8*hi are contiguous in m) -> two 16B stores per lane.
  {
    float* op = out + ((size_t)b * DOUT + (o0 + col)) * MM + m0 + 8 * hi;
    v4f lo = {oacc[0], oacc[1], oacc[2], oacc[3]};
    v4f hi4 = {oacc[4], oacc[5], oacc[6], oacc[7]};
    *(v4f*)op = lo;
    *(v4f*)(op + 4) = hi4;
  }
}

// ---------------------------------------------------------------------------
extern "C" void kernel_launch(void* const* d_in, const int* in_sizes, int n_in,
                              void* d_out, int out_size, void* d_ws, size_t ws_size,
                              hipStream_t stream) {
  const float*     points_xyz = (const float*)d_in[0];
  const float*     features   = (const float*)d_in[1];
  const float*     center_xyz = (const float*)d_in[2];
  const long long* nbr_idx    = (const long long*)d_in[3];  // int64 indices
  const float*     kp         = (const float*)d_in[4];
  const float*     weights    = (const float*)d_in[5];
  float* out   = (float*)d_out;
  float* featT = (float*)d_ws;   // needs B*(N+1)*Din*4 ~= 33.6 MB scratch

  dim3 blk(256);
  kpconv_transpose<<<dim3(BN * (NN / 64 + 1)), blk, 0, stream>>>(features, featT);
  kpconv_main<<<dim3(BN * (MM / 16)), blk, 0, stream>>>(
      points_xyz, center_xyz, nbr_idx, kp, weights, featT, out);
}